// HyperbolicLogisticRegression_70403103916416
// MI455X (gfx1250) — compile-verified
//
#include <hip/hip_runtime.h>
#include <math.h>

// HyperbolicLogisticRegression on gfx1250 (MI455X).
// z:(8,64,256,256) f32, p,w:(16,64) f32 -> logits (8,16,256,256) f32.
// Two K=16 x D=64 GEMMs over M=524288 spatial positions via
// V_WMMA_F32_16X16X4_F32; z streamed from HBM exactly once (memory-bound,
// ~168 MB => ~7 us floor @ 23.3 TB/s); elementwise asinh tail on VALU.

typedef float v2f __attribute__((ext_vector_type(2)));
typedef float v8f __attribute__((ext_vector_type(8)));

#define HWSZ 65536   // H*W
#define DDIM 64
#define KCLS 16
#define NTILES 32768 // (N*H*W)/16

__global__ __launch_bounds__(256) void hyp_logit_wmma_kernel(
    const float* __restrict__ z, const float* __restrict__ p,
    const float* __restrict__ w, float* __restrict__ out)
{
  __shared__ float ph_lds[KCLS * DDIM];  // p_hat = -p
  __shared__ float w_lds[KCLS * DDIM];
  __shared__ float c_dotpw[KCLS];        // p_hat . w
  __shared__ float c_npsq[KCLS];         // |p_hat|^2
  __shared__ float c_wnrm[KCLS];         // |w|

  const int tid = threadIdx.x;

  // Stage p_hat and w into LDS (1024 floats each, 4 per thread).
#pragma unroll
  for (int i = 0; i < 4; ++i) {
    int idx = tid * 4 + i;
    ph_lds[idx] = -p[idx];
    w_lds[idx]  =  w[idx];
  }
  __syncthreads();

  // Per-class scalar constants, one class per thread 0..15.
  if (tid < KCLS) {
    float dpw = 0.f, np = 0.f, wn = 0.f;
#pragma unroll
    for (int d = 0; d < DDIM; ++d) {
      float ph = ph_lds[tid * DDIM + d];
      float ww = w_lds[tid * DDIM + d];
      dpw = fmaf(ph, ww, dpw);
      np  = fmaf(ph, ph, np);
      wn  = fmaf(ww, ww, wn);
    }
    c_dotpw[tid] = dpw;
    c_npsq[tid]  = np;
    c_wnrm[tid]  = sqrtf(wn);
  }
  __syncthreads();

  const int lane = tid & 31;
  const int wid  = tid >> 5;
  const int col  = lane & 15;     // output column within tile
  const int half = lane >> 4;     // 0 | 1
  const int dsub = 2 * half;      // K-subindex base within a 4-chunk

  // Resident A operands: 16 chunks x 2 VGPRs x 2 matrices.
  // A layout (16x4 f32): lane holds row m=col, regs hold K = dsub + {0,1}.
  v2f a_ph[16], a_w[16];
#pragma unroll
  for (int c = 0; c < 16; ++c) {
    int off = col * DDIM + 4 * c + dsub;          // even -> 8B aligned
    a_ph[c] = *(const v2f*)&ph_lds[off];
    a_w[c]  = *(const v2f*)&w_lds[off];
  }

  // Per-class constants for this lane's 8 output rows: k = v + 8*half.
  float dpw_r[8], npsq_r[8], wn_r[8];
#pragma unroll
  for (int v = 0; v < 8; ++v) {
    int k = v + 8 * half;
    dpw_r[v]  = c_dotpw[k];
    npsq_r[v] = c_npsq[k];
    wn_r[v]   = c_wnrm[k];
  }

  const int gwid   = blockIdx.x * 8 + wid;
  const int nwaves = gridDim.x * 8;

  for (int t = gwid; t < NTILES; t += nwaves) {
    const int base = t * 16;          // HW=65536 is a multiple of 16:
    const int n    = base >> 16;      // tile never crosses an n boundary
    const int sp   = base & (HWSZ - 1);
    const float* zb = z + ((size_t)n * DDIM) * HWSZ + sp + col;

    v8f acc_pz = {};
    v8f acc_zw = {};
    float nzp = 0.f;

#pragma unroll
    for (int c = 0; c < 16; ++c) {
      const int d0 = 4 * c + dsub;
      // B layout (4x16 f32): lane holds column `col`, regs K = dsub + {0,1}.
      v2f b;
      b.x = zb[(size_t)d0 * HWSZ];
      b.y = zb[(size_t)(d0 + 1) * HWSZ];
      nzp = fmaf(b.x, b.x, fmaf(b.y, b.y, nzp));
      acc_pz = __builtin_amdgcn_wmma_f32_16x16x4_f32(
          false, a_ph[c], false, b, (short)0, acc_pz, false, false);
      acc_zw = __builtin_amdgcn_wmma_f32_16x16x4_f32(
          false, a_w[c],  false, b, (short)0, acc_zw, false, false);
    }
    // lanes l and l^16 share a column and cover complementary d's
    const float nz = nzp + __shfl_xor(nzp, 16, 32);

    float* ob = out + ((size_t)n * KCLS) * HWSZ + sp + col;
#pragma unroll
    for (int v = 0; v < 8; ++v) {
      const float dpz  = acc_pz[v];
      const float dzw  = acc_zw[v];
      const float npsq = npsq_r[v];
      const float two_dpz_p1 = fmaf(2.f, dpz, 1.f);
      const float denom = fmaf(npsq, nz, two_dpz_p1);    // C=1
      const float rden  = 1.f / (denom + 1e-6f);
      const float alpha = (two_dpz_p1 + nz) * rden;
      const float beta  = (1.f - npsq) * rden;
      const float inner = fmaf(alpha, dpw_r[v], beta * dzw);
      const float mob   = alpha * alpha * npsq
                        + 2.f * alpha * beta * dpz
                        + beta * beta * nz;
      const float num  = 2.f * inner;                    // sqrt(C)=1
      const float den2 = fmaxf(1.f - mob, 1e-6f) * fmaxf(wn_r[v], 1e-6f);
      const float arg  = num / (den2 + 1e-6f);
      const float logit = asinhf(arg) / wn_r[v];         // LAMBDA/(|w|*sqrtC)
      const int k = v + 8 * half;
      ob[(size_t)k * HWSZ] = logit;
    }
  }
}

extern "C" void kernel_launch(void* const* d_in, const int* in_sizes, int n_in,
                              void* d_out, int out_size, void* d_ws, size_t ws_size,
                              hipStream_t stream) {
  const float* z = (const float*)d_in[0];
  const float* p = (const float*)d_in[1];
  const float* w = (const float*)d_in[2];
  float* out = (float*)d_out;
  (void)in_sizes; (void)n_in; (void)out_size; (void)d_ws; (void)ws_size;
  // 1024 blocks x 8 waves = 8192 waves; 4 tiles per wave.
  hipLaunchKernelGGL(hyp_logit_wmma_kernel, dim3(1024), dim3(256), 0, stream,
                     z, p, w, out);
}